// GraphConvolution_39204461478459
// MI455X (gfx1250) — compile-verified
//
#include <hip/hip_runtime.h>
#include <cmath>

// GraphConvolution on MI455X (gfx1250, wave32).
// Bandwidth-bound (~819 MB output stores dominate). The two real GEMMs
// (x@core_w^T K=512 and avg@W_out^T K=4) run on V_WMMA_F32_16X16X4_F32.

typedef __attribute__((ext_vector_type(2))) float v2f;
typedef __attribute__((ext_vector_type(4))) float v4f;
typedef __attribute__((ext_vector_type(8))) float v8f;

#define F_IN   512
#define F_OUT  64
#define NCORES 4

// ---------------------------------------------------------------- zero ws ---
__global__ __launch_bounds__(256) void gc_zero(float* __restrict__ p, int n) {
  int i = blockIdx.x * 256 + threadIdx.x;
  if (i < n) p[i] = 0.0f;
}

// --------------------------------------------------- s = tanh(x @ Wc^T + b) --
// One wave = one 16-node tile. f32 WMMA 16x16x4:
//   A (16x4):  lanes 0-15 hold M=lane, K={0,1}; lanes 16-31 hold M=lane-16, K={2,3}
//   B (4x16):  mirrored; B column n == core index. Columns 4..15 of C are dead,
//              so lanes with col>=4 read duplicate (finite) B data — NO masking,
//              NO divergence in the inner loop (EXEC stays all-1s throughout).
//   C (16x16): VGPR j -> M = j (lanes 0-15) / j+8 (lanes 16-31), N = lane&15
__global__ __launch_bounds__(256) void gc_signal_wmma(
    const float* __restrict__ x, const float* __restrict__ core_w,
    const float* __restrict__ core_b, float* __restrict__ s,
    int nNodes, int nTiles) {
  // Stage core_w in LDS as bmat[kstep][core][kk] : 128*4*4 floats = 8 KB.
  __shared__ __align__(16) float bmat[128 * 16];
  for (int i = threadIdx.x; i < 128 * 16; i += 256) {
    int step = i >> 4, col = (i >> 2) & 3, kk = i & 3;
    bmat[i] = core_w[col * F_IN + step * 4 + kk];
  }
  __syncthreads();

  int wave = (int)((blockIdx.x * blockDim.x + threadIdx.x) >> 5);
  if (wave >= nTiles) return;                // wave-uniform branch
  int lane = threadIdx.x & 31;
  int row  = lane & 15;
  int kof  = (lane >> 4) << 1;               // 0 or 2
  int node = wave * 16 + row;
  int nodeC = node < nNodes ? node : (nNodes - 1);        // clamp tail loads
  const float* xr = x + (long)nodeC * F_IN + kof;
  int col  = row & 3;                        // clamped B column (cols>=4 dead)
  const float* bp = bmat + col * 4 + kof;    // LDS base for this lane

  v8f acc0 = {}, acc1 = {};                  // 2 chains: ILP across WMMAs
  #pragma unroll 4
  for (int k = 0; k < F_IN; k += 8) {
    v2f a0 = *reinterpret_cast<const v2f*>(xr + k);
    v2f a1 = *reinterpret_cast<const v2f*>(xr + k + 4);
    v2f b0 = *reinterpret_cast<const v2f*>(bp + (k << 1));        // step k/8
    v2f b1 = *reinterpret_cast<const v2f*>(bp + (k << 1) + 16);   // step k/8+1
    acc0 = __builtin_amdgcn_wmma_f32_16x16x4_f32(false, a0, false, b0,
                                                 (short)0, acc0, false, false);
    acc1 = __builtin_amdgcn_wmma_f32_16x16x4_f32(false, a1, false, b1,
                                                 (short)0, acc1, false, false);
  }
  v8f acc = acc0 + acc1;

  if (row < NCORES) {                        // only C columns 0..3 are real
    float bias = core_b[row];
    int base = wave * 16 + ((lane < 16) ? 0 : 8);
    if (wave * 16 + 16 <= nNodes) {          // full tile (always, N%16==0)
      #pragma unroll
      for (int j = 0; j < 8; ++j)
        s[(long)(base + j) * NCORES + row] = tanhf(acc[j] + bias);
    } else {
      #pragma unroll
      for (int j = 0; j < 8; ++j)
        if (base + j < nNodes)
          s[(long)(base + j) * NCORES + row] = tanhf(acc[j] + bias);
    }
  }
}

// ------------------------------------------------ edge scatter (mean numer) --
__global__ __launch_bounds__(256) void gc_scatter(
    const int* __restrict__ src, const int* __restrict__ tgt,
    const float* __restrict__ s, float* __restrict__ seg,
    float* __restrict__ cnt, int E) {
  int e = blockIdx.x * 256 + threadIdx.x;
  if (e >= E) return;
  int u = src[e], v = tgt[e];
  v4f sv = *reinterpret_cast<const v4f*>(s + (long)u * NCORES);
  float* segv = seg + (long)v * NCORES;
  atomicAdd(segv + 0, sv.x);
  atomicAdd(segv + 1, sv.y);
  atomicAdd(segv + 2, sv.z);
  atomicAdd(segv + 3, sv.w);
  atomicAdd(cnt + v, 1.0f);
}

// -------------------------------------- t[n,o] = sum_c avg[n,c]*W_out[o,c] ---
// Exact-fit K=4 WMMA: A = avg tile (16x4), B = W_out^T (4x16), 4 N-tiles.
// All 64 B columns are valid -> no masking anywhere.
__global__ __launch_bounds__(256) void gc_tmat_wmma(
    const float* __restrict__ seg, const float* __restrict__ cnt,
    const float* __restrict__ W_out, float* __restrict__ t,
    int nNodes, int nTiles) {
  int wave = (int)((blockIdx.x * blockDim.x + threadIdx.x) >> 5);
  if (wave >= nTiles) return;
  int lane = threadIdx.x & 31;
  int row  = lane & 15;
  int kof  = (lane >> 4) << 1;
  int node  = wave * 16 + row;
  int nodeC = node < nNodes ? node : (nNodes - 1);

  float c   = cnt[nodeC];
  float inv = (c > 0.0f) ? (1.0f / fmaxf(c, 1.0f)) : 0.0f;  // matches jnp.where
  v2f av = *reinterpret_cast<const v2f*>(seg + (long)nodeC * NCORES + kof);
  v2f a;  a.x = av.x * inv;  a.y = av.y * inv;

  bool full = (wave * 16 + 16 <= nNodes);
  int mbase = wave * 16 + ((lane < 16) ? 0 : 8);
  #pragma unroll
  for (int nt = 0; nt < 4; ++nt) {
    int o = nt * 16 + row;                        // output column (0..63)
    v2f b = *reinterpret_cast<const v2f*>(W_out + (long)o * NCORES + kof);
    v8f acc = {};
    acc = __builtin_amdgcn_wmma_f32_16x16x4_f32(false, a, false, b,
                                                (short)0, acc, false, false);
    if (full) {
      #pragma unroll
      for (int j = 0; j < 8; ++j)
        t[(long)(mbase + j) * F_OUT + o] = acc[j];
    } else {
      #pragma unroll
      for (int j = 0; j < 8; ++j)
        if (mbase + j < nNodes) t[(long)(mbase + j) * F_OUT + o] = acc[j];
    }
  }
}

// ------------------- out[n,f,o] = relu(x[n,f]*t[n,o] + b_out[o]), streamed ---
// Block per node; 16 KB per node via coalesced NON-TEMPORAL b128 stores
// (output is write-once, 819 MB >> 192 MB L2 -- keep x resident instead).
__global__ __launch_bounds__(256) void gc_outer(
    const float* __restrict__ x, const float* __restrict__ t,
    const float* __restrict__ b_out, float* __restrict__ out) {
  int n   = blockIdx.x;
  int tid = threadIdx.x;
  __shared__ float xs[F_OUT], ts[F_OUT], bs[F_OUT];
  if (tid < F_OUT) {
    xs[tid] = x[(long)n * F_IN + tid];      // only first F_OUT features used
    ts[tid] = t[(long)n * F_OUT + tid];
    bs[tid] = b_out[tid];
  }
  __syncthreads();
  float* ob = out + (long)n * (F_OUT * F_OUT);
  #pragma unroll
  for (int i = 0; i < 4; ++i) {
    int cidx = tid + i * 256;               // float4 chunk 0..1023
    int f = cidx >> 4;                      // feature row
    int o = (cidx & 15) << 2;               // output col base
    float xf = xs[f];
    v4f v;
    v.x = fmaxf(fmaf(xf, ts[o + 0], bs[o + 0]), 0.0f);
    v.y = fmaxf(fmaf(xf, ts[o + 1], bs[o + 1]), 0.0f);
    v.z = fmaxf(fmaf(xf, ts[o + 2], bs[o + 2]), 0.0f);
    v.w = fmaxf(fmaf(xf, ts[o + 3], bs[o + 3]), 0.0f);
    __builtin_nontemporal_store(v, reinterpret_cast<v4f*>(ob + (long)cidx * 4));
  }
}

// ---------------------------------------------------------------------------
extern "C" void kernel_launch(void* const* d_in, const int* in_sizes, int n_in,
                              void* d_out, int out_size, void* d_ws, size_t ws_size,
                              hipStream_t stream) {
  const float* x      = (const float*)d_in[0];
  const int*   ei     = (const int*)  d_in[1];
  const float* core_w = (const float*)d_in[2];
  const float* core_b = (const float*)d_in[3];
  const float* W_out  = (const float*)d_in[4];
  const float* b_out  = (const float*)d_in[5];
  float*       out    = (float*)d_out;

  int N = in_sizes[0] / F_IN;     // 50000
  int E = in_sizes[1] / 2;        // 1600000
  const int* src = ei;            // edge_index[0]
  const int* tgt = ei + E;        // edge_index[1]

  // Workspace layout (floats): s[N*4] | seg[N*4] | cnt[N] | t[N*64]  (~14.6 MB)
  float* ws  = (float*)d_ws;
  float* s   = ws;
  float* seg = ws + (size_t)N * 4;
  float* cnt = ws + (size_t)N * 8;
  float* t   = ws + (size_t)N * 9;

  int nTiles = (N + 15) / 16;
  int tileBlocks = (nTiles + 7) / 8;      // 8 waves (256 thr) per block

  gc_zero<<<(N * 5 + 255) / 256, 256, 0, stream>>>(seg, N * 5);
  gc_signal_wmma<<<tileBlocks, 256, 0, stream>>>(x, core_w, core_b, s, N, nTiles);
  gc_scatter<<<(E + 255) / 256, 256, 0, stream>>>(src, tgt, s, seg, cnt, E);
  gc_tmat_wmma<<<tileBlocks, 256, 0, stream>>>(seg, cnt, W_out, t, N, nTiles);
  gc_outer<<<N, 256, 0, stream>>>(x, t, b_out, out);
}